// MLDEL_2_52269751992447
// MI455X (gfx1250) — compile-verified
//
#include <hip/hip_runtime.h>
#include <hip/hip_bf16.h>

typedef __attribute__((ext_vector_type(2))) float v2f;
typedef __attribute__((ext_vector_type(8))) float v8f;

#define MINN 1e-15f
#define FEPS 1e-7f

// ---------- helpers ----------
__device__ __forceinline__ float wave_sum32(float v){
#pragma unroll
  for (int o = 16; o > 0; o >>= 1) v += __shfl_xor(v, o, 32);
  return v;
}
__device__ __forceinline__ float dot4(const float4& a){
  return a.x*a.x + a.y*a.y + a.z*a.z + a.w*a.w;
}
__device__ __forceinline__ float4 scl4(const float4& a, float s){
  return make_float4(s*a.x, s*a.y, s*a.z, s*a.w);
}
__device__ __forceinline__ float4 fma4(const float4& a, float s, const float4& b){
  return make_float4(s*a.x + b.x, s*a.y + b.y, s*a.z + b.z, s*a.w + b.w);
}

// scale factor of logmap0(proj(expmap0(u))) given ||u[1:]||^2 (reference clamp chain)
__device__ __forceinline__ float rt_scale(float r2, float sqrtK, float K){
  float r     = fmaxf(sqrtf(r2), MINN);
  float th    = r / sqrtK;
  float sh    = sinhf(th);
  float restn = sqrtK * sh;                               // ||rest||
  float t     = sqrtf(fmaxf(K + restn*restn, FEPS));      // proj clip
  float th2   = acoshf(fmaxf(t / sqrtK, 1.0f + FEPS));    // logmap clip
  return (sqrtK * th2 / fmaxf(restn, MINN)) * (sqrtK * sh / r);
}

// ---------- setup: c, sqrtK, bias tangents ----------
__global__ void k_setup(const float* raw_c, const float* b1, const float* b2, const float* b3,
                        float* scal, float* b1t, float* b2t, float* b3t, float* c_out){
  int lane = threadIdx.x & 31;
  float rc = raw_c[0];
  float c  = log1pf(expf(rc)) + 1e-5f;
  float K  = 1.0f / c;
  float sqrtK = sqrtf(K);
  if (lane == 0){ scal[0] = c; scal[1] = sqrtK; scal[2] = K; c_out[0] = c; }
  const float* bs[3] = {b1, b2, b3};
  float*       bt[3] = {b1t, b2t, b3t};
  for (int i = 0; i < 3; ++i){
    float4 u = *(const float4*)(bs[i] + lane*4);
    if (lane == 0) u.x = 0.0f;                 // proj_tan0
    float r2 = wave_sum32(dot4(u));
    float sc = rt_scale(r2, sqrtK, K);
    *(float4*)(bt[i] + lane*4) = scl4(u, sc);
  }
}

// ---------- weight packing: P[((k>>1)*N + n)*2 + (k&1)] = (k? W[k,n] : 0) ----------
__global__ void k_packW(const float* __restrict__ W, float* __restrict__ P, int Kdim, int Ncols){
  int idx = blockIdx.x * 256 + threadIdx.x;
  if (idx >= Kdim * Ncols) return;
  int k = idx / Ncols, n = idx - k * Ncols;
  float v = (k > 0) ? W[(size_t)k * Ncols + n] : 0.0f;
  P[((size_t)(k >> 1) * Ncols + n) * 2 + (k & 1)] = v;
}

// ---------- alpha_i = RT scale of [0, A1_row[1:]] ----------
__global__ void k_alpha(const float* __restrict__ A, float* __restrict__ alpha,
                        const float* scal, int N){
  int node = blockIdx.x * 8 + (threadIdx.x >> 5);
  int lane = threadIdx.x & 31;
  if (node >= N) return;
  float sqrtK = scal[1], K = scal[2];
  const float* row = A + (size_t)node * 256;
  float acc = 0.0f;
#pragma unroll
  for (int j = 0; j < 2; ++j){
    float4 u = *(const float4*)(row + lane*8 + j*4);
    if (lane == 0 && j == 0) u.x = 0.0f;       // proj_tan0
    acc += dot4(u);
  }
  float r2 = wave_sum32(acc);
  if (lane == 0) alpha[node] = rt_scale(r2, sqrtK, K);
}

// ---------- f32 WMMA GEMM: C[M,Ncols] = A[M,Kdim] @ W, 1 wave -> 16x64 strip ----------
__global__ void __launch_bounds__(256)
k_gemm_wmma(const float* __restrict__ A, const float* __restrict__ P,
            float* __restrict__ C, int Mtiles, int ldA, int Kdim, int Ncols){
  const int wid = blockIdx.x * 8 + (threadIdx.x >> 5);
  const int NG  = Ncols >> 6;
  if (wid >= Mtiles * NG) return;                 // wave-uniform (EXEC all-1s for WMMA)
  const int mt   = wid / NG;
  const int ng   = wid - mt * NG;
  const int lane = threadIdx.x & 31;
  const int half = lane >> 4;
  const int lm   = lane & 15;
  const int n0   = ng * 64;

  v8f acc0 = {}, acc1 = {}, acc2 = {}, acc3 = {};
  const float* ap = A + (size_t)(mt*16 + lm) * ldA + (half << 1);
  const v2f*   pb = (const v2f*)P;

  for (int k0 = 0; k0 < Kdim; k0 += 4){
    v2f a = *(const v2f*)(ap + k0);               // lanes0-15:(K0,K1) lanes16-31:(K2,K3)
    const v2f* bb = pb + (size_t)((k0 >> 1) + half) * Ncols + n0 + lm;
    v2f b0 = bb[0];
    v2f b1 = bb[16];
    v2f b2 = bb[32];
    v2f b3 = bb[48];
    acc0 = __builtin_amdgcn_wmma_f32_16x16x4_f32(false, a, false, b0, (short)0, acc0, false, false);
    acc1 = __builtin_amdgcn_wmma_f32_16x16x4_f32(false, a, false, b1, (short)0, acc1, false, false);
    acc2 = __builtin_amdgcn_wmma_f32_16x16x4_f32(false, a, false, b2, (short)0, acc2, false, false);
    acc3 = __builtin_amdgcn_wmma_f32_16x16x4_f32(false, a, false, b3, (short)0, acc3, false, false);
  }
  float* c0 = C + (size_t)(mt*16 + half*8) * Ncols + n0 + lm;
#pragma unroll
  for (int j = 0; j < 8; ++j){
    float* cr = c0 + (size_t)j * Ncols;           // row M = mt*16 + half*8 + j
    cr[0]  = acc0[j];
    cr[16] = acc1[j];
    cr[32] = acc2[j];
    cr[48] = acc3[j];
  }
}

// ---------- EW1a: m1 -> a1full (Y), La2 (in-place X) ----------
__global__ void k_ew1a(float* __restrict__ m1, const float* __restrict__ alpha,
                       const float* __restrict__ nvec, const float* scal,
                       const float* __restrict__ b1t, float* __restrict__ a1full, int N){
  int node = blockIdx.x * 8 + (threadIdx.x >> 5);
  int lane = threadIdx.x & 31;
  if (node >= N) return;
  float sqrtK = scal[1], K = scal[2];
  float al = alpha[node], nn = nvec[node];
  size_t ro = (size_t)node * 128 + lane*4;
  float4 u = scl4(*(const float4*)(m1 + ro), al);
  if (lane == 0) u.x = 0.0f;
  float4 bt = *(const float4*)(b1t + lane*4);
  float  sc = rt_scale(wave_sum32(dot4(u)), sqrtK, K);
  float4 s1 = fma4(u, sc, bt);                    // tangent, comp0 = 0
  float  r2s = wave_sum32(dot4(s1));
  // expmap0(s1)
  float r  = fmaxf(sqrtf(r2s), MINN);
  float th = r / sqrtK;
  float sh = sinhf(th);
  float se = sqrtK * sh / r;
  float restn = sqrtK * sh;
  float t  = sqrtf(fmaxf(K + restn*restn, FEPS));
  float4 rest = scl4(s1, se);
  float4 a1o = rest; if (lane == 0) a1o.x = t;
  *(float4*)(a1full + ro) = a1o;
  // logmap0
  float sl = sqrtK * acoshf(fmaxf(t / sqrtK, 1.0f + FEPS)) / fmaxf(restn, MINN);
  float4 w = scl4(rest, sl * nn);                 // n * La1
  float sc2 = rt_scale(wave_sum32(dot4(w)), sqrtK, K);
  *(float4*)(m1 + ro) = scl4(w, sc2);             // La2
}

// ---------- EW1b / EW3: t = RT(alpha*m + ... ) with bias ----------
__global__ void k_ew_bias(float* __restrict__ m, const float* __restrict__ alpha,
                          const float* scal, const float* __restrict__ bt4, int N){
  int node = blockIdx.x * 8 + (threadIdx.x >> 5);
  int lane = threadIdx.x & 31;
  if (node >= N) return;
  float sqrtK = scal[1], K = scal[2];
  float al = alpha ? alpha[node] : 1.0f;
  size_t ro = (size_t)node * 128 + lane*4;
  float4 u = scl4(*(const float4*)(m + ro), al);
  if (lane == 0) u.x = 0.0f;
  float4 bt = *(const float4*)(bt4 + lane*4);
  float  sc = rt_scale(wave_sum32(dot4(u)), sqrtK, K);
  float4 s  = fma4(u, sc, bt);
  float  sc3 = rt_scale(wave_sum32(dot4(s)), sqrtK, K);
  *(float4*)(m + ro) = scl4(s, sc3);
}

// ---------- EW2: sp2 -> Lx1 = RT(RT((1-n)*RT(sp2)) + La2), in place ----------
__global__ void k_ew2(float* __restrict__ sp, const float* __restrict__ La2,
                      const float* __restrict__ nvec, const float* scal, int N){
  int node = blockIdx.x * 8 + (threadIdx.x >> 5);
  int lane = threadIdx.x & 31;
  if (node >= N) return;
  float sqrtK = scal[1], K = scal[2];
  float nn = nvec[node];
  size_t ro = (size_t)node * 128 + lane*4;
  float4 g = *(const float4*)(sp + ro);
  if (lane == 0) g.x = 0.0f;
  float scg = rt_scale(wave_sum32(dot4(g)), sqrtK, K);
  float4 v  = scl4(g, scg * (1.0f - nn));
  float scv = rt_scale(wave_sum32(dot4(v)), sqrtK, K);
  float4 l2 = *(const float4*)(La2 + ro);
  float4 q  = fma4(v, scv, l2);
  float scq = rt_scale(wave_sum32(dot4(q)), sqrtK, K);
  *(float4*)(sp + ro) = scl4(q, scq);
}

// ---------- zero ----------
__global__ void k_zero(float* p, size_t n){
  size_t i = (size_t)blockIdx.x * blockDim.x + threadIdx.x;
  if (i < n) p[i] = 0.0f;
}

// ---------- SPMM scatter: y[row] += val * x[col], 32 lanes per edge ----------
__global__ void k_spmm(const int* __restrict__ rows, const int* __restrict__ cols,
                       const float* __restrict__ vals, const float* __restrict__ x,
                       float* __restrict__ y, int E){
  long tid = (long)blockIdx.x * blockDim.x + threadIdx.x;
  int e = (int)(tid >> 5);
  if (e >= E) return;
  int c4 = ((int)tid & 31) * 4;
  int rr = rows[e], cc = cols[e];
  float v = vals[e];
  float4 xv = *(const float4*)(x + (size_t)cc * 128 + c4);
  float* yp = y + (size_t)rr * 128 + c4;
  atomicAdd(yp + 0, v * xv.x);
  atomicAdd(yp + 1, v * xv.y);
  atomicAdd(yp + 2, v * xv.z);
  atomicAdd(yp + 3, v * xv.w);
}

// ---------- final: out = expmap0([RT(sp3) | a1full]) ----------
__global__ void k_final(const float* __restrict__ sp3, const float* __restrict__ a1full,
                        const float* scal, float* __restrict__ out, int N){
  int node = blockIdx.x * 8 + (threadIdx.x >> 5);
  int lane = threadIdx.x & 31;
  if (node >= N) return;
  float sqrtK = scal[1], K = scal[2];
  size_t ro = (size_t)node * 128 + lane*4;
  float4 z4 = *(const float4*)(sp3 + ro);
  if (lane == 0) z4.x = 0.0f;
  float scz = rt_scale(wave_sum32(dot4(z4)), sqrtK, K);
  float4 z  = scl4(z4, scz);
  float4 a4 = *(const float4*)(a1full + ro);
  float r2c = wave_sum32(dot4(z) + dot4(a4));      // norm over comps 1..255
  float r  = fmaxf(sqrtf(r2c), MINN);
  float th = r / sqrtK;
  float sh = sinhf(th);
  float se = sqrtK * sh / r;
  float restn = sqrtK * sh;
  float t  = sqrtf(fmaxf(K + restn*restn, FEPS));
  float4 o1 = scl4(z, se);  if (lane == 0) o1.x = t;
  float4 o2 = scl4(a4, se);
  float* orow = out + (size_t)node * 256;
  *(float4*)(orow + lane*4)       = o1;
  *(float4*)(orow + 128 + lane*4) = o2;
}

extern "C" void kernel_launch(void* const* d_in, const int* in_sizes, int n_in,
                              void* d_out, int out_size, void* d_ws, size_t ws_size,
                              hipStream_t stream){
  const float* A1    = (const float*)d_in[0];
  const int*   rows  = (const int*)  d_in[1];
  const int*   cols  = (const int*)  d_in[2];
  const float* vals  = (const float*)d_in[3];
  const float* rawc  = (const float*)d_in[4];
  const float* lin1  = (const float*)d_in[5];
  const float* lin1b = (const float*)d_in[6];
  const float* nvec  = (const float*)d_in[7];
  const float* g1w   = (const float*)d_in[8];
  const float* g1b   = (const float*)d_in[9];
  const float* g2w   = (const float*)d_in[10];
  const float* g2b   = (const float*)d_in[11];
  float* out = (float*)d_out;
  const int N = in_sizes[7];          // 100000
  const int E = in_sizes[1];          // 1600000

  float* ws    = (float*)d_ws;
  float* scal  = ws;                  // 16 floats
  float* b1t   = ws + 16;             // 128
  float* bg1t  = ws + 16 + 128;       // 128
  float* bg2t  = ws + 16 + 256;       // 128
  float* alpha = ws + 512;            // N
  float* P1a   = ws + 131072;         // Lin1 packed: 256*128
  float* P1b   = P1a + 32768;         // gc1 packed
  float* P2    = P1b + 32768;         // gc2 packed: 128*128
  const size_t stride = (size_t)N * 128;
  float* W = ws + 262144;             // big buffers (4 x N*128)
  float* X = W + stride;
  float* Y = X + stride;
  float* Z = Y + stride;

  const int nbNode = (N + 7) / 8;                       // wave-per-node, 8 nodes/block
  const int Mtiles = N / 16;
  const int nbGemm = (Mtiles * 2 + 7) / 8;              // NG = 128/64 = 2
  const int nbZero = (int)((stride + 255) / 256);
  const int nbEdge = (int)(((long)E * 32 + 255) / 256);

  k_setup<<<1, 32, 0, stream>>>(rawc, lin1b, g1b, g2b, scal, b1t, bg1t, bg2t,
                                out + (size_t)N * 256);
  k_packW<<<(256*128)/256, 256, 0, stream>>>(lin1, P1a, 256, 128);
  k_packW<<<(256*128)/256, 256, 0, stream>>>(g1w,  P1b, 256, 128);
  k_packW<<<(128*128)/256, 256, 0, stream>>>(g2w,  P2,  128, 128);
  k_alpha<<<nbNode, 256, 0, stream>>>(A1, alpha, scal, N);

  // m1 = A1 @ Lin1 -> X ; then a1full -> Y, La2 -> X
  k_gemm_wmma<<<nbGemm, 256, 0, stream>>>(A1, P1a, X, Mtiles, 256, 256, 128);
  k_ew1a<<<nbNode, 256, 0, stream>>>(X, alpha, nvec, scal, b1t, Y, N);

  // m2 = A1 @ gc1 -> Z ; then t2 in place
  k_gemm_wmma<<<nbGemm, 256, 0, stream>>>(A1, P1b, Z, Mtiles, 256, 256, 128);
  k_ew_bias<<<nbNode, 256, 0, stream>>>(Z, alpha, scal, bg1t, N);

  // sp2 = spmm(t2) -> W ; then Lx1 in place
  k_zero<<<nbZero, 256, 0, stream>>>(W, stride);
  k_spmm<<<nbEdge, 256, 0, stream>>>(rows, cols, vals, Z, W, E);
  k_ew2<<<nbNode, 256, 0, stream>>>(W, X, nvec, scal, N);

  // m3 = Lx1 @ gc2 -> Z ; then t3 in place
  k_gemm_wmma<<<nbGemm, 256, 0, stream>>>(W, P2, Z, Mtiles, 128, 128, 128);
  k_ew_bias<<<nbNode, 256, 0, stream>>>(Z, (const float*)nullptr, scal, bg2t, N);

  // sp3 = spmm(t3) -> X ; final output
  k_zero<<<nbZero, 256, 0, stream>>>(X, stride);
  k_spmm<<<nbEdge, 256, 0, stream>>>(rows, cols, vals, Z, X, E);
  k_final<<<nbNode, 256, 0, stream>>>(X, Y, scal, out, N);
}